// YoloLoss_62242666053671
// MI455X (gfx1250) — compile-verified
//
#include <hip/hip_runtime.h>
#include <math.h>

#define NB_CH 255
#define NA 3
#define NH 52
#define NW 52
#define HW (NH*NW)
#define NC 80
#define TMAX 128
#define IGNORE_THRESH 0.6f

// Anchors / STRIDE(=8); first NA are the masked ones (mask = [0,1,2] -> identity LUT)
__device__ __constant__ float c_aw[9] = {1.25f, 2.0f, 4.125f, 3.75f, 7.75f, 7.375f, 14.5f, 19.5f, 46.625f};
__device__ __constant__ float c_ah[9] = {1.625f, 3.75f, 2.875f, 7.625f, 5.625f, 14.875f, 11.25f, 24.75f, 40.75f};

typedef float v2f __attribute__((ext_vector_type(2)));
typedef float v8f __attribute__((ext_vector_type(8)));
typedef unsigned int u32x4 __attribute__((ext_vector_type(4)));
typedef int i32x4 __attribute__((ext_vector_type(4)));
typedef int i32x8 __attribute__((ext_vector_type(8)));

#if __has_builtin(__builtin_amdgcn_tensor_load_to_lds) && __has_builtin(__builtin_amdgcn_s_wait_tensorcnt)
#define USE_TDM 1
#endif

__device__ __forceinline__ float softplusf(float x) {
    return fmaxf(x, 0.0f) + log1pf(expf(-fabsf(x)));
}

// Reduce 256 per-thread partials to one scalar (valid in lane/tid 0).
// Uses V_WMMA_F32_16X16X4_F32 on wave 0: D = A*ones + C, so every D entry is a
// row-sum of A; column 0 of D (lane0 M=0..7, lane16 M=8..15) sums to total(A).
// The LDS->A mapping is an arbitrary bijection, which is fine since B==ones.
__device__ __forceinline__ float block_reduce_256(float val, float* sRed, int tid) {
    sRed[tid] = val;
    __syncthreads();
    float total = 0.0f;
    if (tid < 32) {            // uniform for wave 0 -> EXEC all ones inside
#if __has_builtin(__builtin_amdgcn_wmma_f32_16x16x4_f32)
        v8f acc = {};
        v2f ones = {1.0f, 1.0f};
        for (int c = 0; c < 4; ++c) {
            v2f av;
            av[0] = sRed[c * 64 + tid];
            av[1] = sRed[c * 64 + 32 + tid];
            acc = __builtin_amdgcn_wmma_f32_16x16x4_f32(
                false, av, false, ones, (short)0, acc, false, false);
        }
        float s = acc[0] + acc[1] + acc[2] + acc[3] + acc[4] + acc[5] + acc[6] + acc[7];
        total = s + __shfl(s, 16, 32);   // col0: M0..7 in lane0 + M8..15 in lane16
#else
        float s = sRed[tid] + sRed[tid + 32] + sRed[tid + 64] + sRed[tid + 96]
                + sRed[tid + 128] + sRed[tid + 160] + sRed[tid + 192] + sRed[tid + 224];
        for (int off = 16; off > 0; off >>= 1) s += __shfl_down(s, off, 32);
        total = s;
#endif
    }
    return total;
}

__global__ __launch_bounds__(256) void k_init(int* __restrict__ cellTarg, int n) {
    int idx = blockIdx.x * 256 + threadIdx.x;
    if (idx < n) cellTarg[idx] = -1;
}

// Per-batch target prep: validity prefix, best-of-9 anchor, scatter (last t wins
// via atomicMax), and per-target regression targets.
__global__ __launch_bounds__(256) void k_prep(const float* __restrict__ tgt, int T,
                                              int* __restrict__ cellTarg,
                                              int* __restrict__ nValid,
                                              float* __restrict__ gtBox,
                                              float* __restrict__ targData) {
    int b = blockIdx.x;
    int t = threadIdx.x;
    __shared__ int sV;
    const float* tb = tgt + (size_t)b * T * 5;
    if (t == 0) {
        int V = T;
        for (int k = 0; k < T; ++k) {
            if (tb[k * 5 + 1] == 0.0f) { V = k; break; }   // cumprod(tx!=0) prefix
        }
        nValid[b] = V;
        sV = V;
    }
    __syncthreads();
    if (t >= T) return;
    float cls = tb[t * 5 + 0];
    float gx = tb[t * 5 + 1] * (float)NW;
    float gy = tb[t * 5 + 2] * (float)NH;
    float gw = tb[t * 5 + 3] * (float)NW;
    float gh = tb[t * 5 + 4] * (float)NH;
    float* gb = gtBox + ((size_t)b * T + t) * 4;
    gb[0] = gx; gb[1] = gy; gb[2] = gw; gb[3] = gh;
    bool valid = t < sV;
    // argmax of anchor-overlap IoU (first index wins ties, like jnp.argmax)
    float best = -1.0f; int bn = 0;
    for (int n = 0; n < 9; ++n) {
        float aw = c_aw[n], ah = c_ah[n];
        float inter = fminf(gw, aw) * fminf(gh, ah);
        float uni = gw * gh + aw * ah - inter;
        float r = inter / uni;
        if (r > best) { best = r; bn = n; }
    }
    if (valid && bn < NA) {                 // sel: LUT[mask 0..2] identity
        int gi = (int)gx;                   // trunc, matches astype(int32)
        int gj = (int)gy;
        if (gi >= 0 && gi < NW && gj >= 0 && gj < NH) {   // mode='drop'
            int cell = ((b * NA + bn) * NH + gj) * NW + gi;
            atomicMax(&cellTarg[cell], t);  // last target (max t) wins
            float aw = c_aw[bn], ah = c_ah[bn];
            float* td = targData + ((size_t)b * T + t) * 8;
            td[0] = gx - (float)gi;
            td[1] = gy - (float)gj;
            td[2] = logf(gw / aw);
            td[3] = logf(gh / ah);
            td[4] = 2.0f - gw * gh / (float)(NH * NW);    // coord_scale
            td[5] = cls;
        }
    }
}

// Main pass: one thread per (b,a,j,i) cell. GT boxes of this batch staged into
// LDS by the Tensor Data Mover (TENSOR_LOAD_TO_LDS + s_wait_tensorcnt).
__global__ __launch_bounds__(256) void k_main(const float* __restrict__ inp, int T,
                                              const int* __restrict__ cellTarg,
                                              const int* __restrict__ nValid,
                                              const float* __restrict__ gtBox,
                                              const float* __restrict__ targData,
                                              float* __restrict__ partials) {
    __shared__ float sBox[TMAX * 4];
    __shared__ int sV;
    __shared__ float sRed[256];
    const int blocksPerB = (NA * HW + 255) / 256;
    int b = blockIdx.x / blocksPerB;
    int chunk = blockIdx.x - b * blocksPerB;
    int tid = threadIdx.x;
    int Tc = (T < TMAX) ? T : TMAX;
    if (tid == 0) sV = nValid[b];
#ifdef USE_TDM
    if (tid < 32) {   // wave 0 only: TDM ignores EXEC, avoid 8 duplicate DMAs
        // D# for a 2D row-tensor: data_size=4B, tile = (T*4) x 1 elements.
        unsigned int nElem = (unsigned int)(Tc * 4);
        unsigned long long ga =
            (unsigned long long)(const void*)(gtBox + (size_t)b * T * 4);
        unsigned int ldsAddr = (unsigned int)(size_t)(&sBox[0]);
        u32x4 g0 = { 1u,                                     // count=1 valid D#
                     ldsAddr,                                // lds_addr[63:32]
                     (unsigned int)(ga & 0xFFFFFFFFu),       // global_addr lo
                     ((unsigned int)((ga >> 32) & 0x01FFFFFFu)) | (2u << 30) }; // +type=2
        i32x8 g1 = { (int)(2u << 16),       // data_size=2 (4 bytes), mask=0
                     (int)(nElem << 16),    // tensor_dim0[15:0]
                     (int)(1u << 16),       // tensor_dim0[31:16]=0, tensor_dim1=1
                     (int)(nElem << 16),    // tensor_dim1[31:16]=0, tile_dim0
                     1,                     // tile_dim1=1, tile_dim2=0
                     (int)nElem,            // tensor_dim0_stride[31:0]
                     0,                     // stride hi, tensor_dim1_stride lo
                     0 };
        i32x4 z4 = { 0, 0, 0, 0 };
#if __clang_major__ >= 23
        i32x8 z8 = { 0, 0, 0, 0, 0, 0, 0, 0 };
        __builtin_amdgcn_tensor_load_to_lds(g0, g1, z4, z4, z8, 0);
#else
        __builtin_amdgcn_tensor_load_to_lds(g0, g1, z4, z4, 0);
#endif
        __builtin_amdgcn_s_wait_tensorcnt(0);
    }
#else
    for (int k = tid; k < Tc * 4; k += 256) sBox[k] = gtBox[(size_t)b * T * 4 + k];
#endif
    __syncthreads();
    int V = sV;
    if (V > Tc) V = Tc;
    int cidx = chunk * 256 + tid;
    float loss = 0.0f;
    if (cidx < NA * HW) {
        int a = cidx / HW;
        int rem = cidx - a * HW;
        int j = rem / NW;
        int i = rem - j * NW;
        const float* base = inp + ((size_t)b * NB_CH + a * 85) * HW + rem;
        float lx = base[0];
        float ly = base[HW];
        float lw = base[2 * HW];
        float lh = base[3 * HW];
        float lconf = base[4 * HW];
        float aw = c_aw[a], ah = c_ah[a];
        float px = (float)i + 1.0f / (1.0f + expf(-lx));
        float py = (float)j + 1.0f / (1.0f + expf(-ly));
        float pw = expf(lw) * aw;
        float ph = expf(lh) * ah;
        float pa = pw * ph;
        float cur = 0.0f;
        for (int t = 0; t < V; ++t) {
            float gx = sBox[t * 4 + 0], gy = sBox[t * 4 + 1];
            float gw = sBox[t * 4 + 2], gh = sBox[t * 4 + 3];
            float mx = fminf(px - 0.5f * pw, gx - 0.5f * gw);
            float Mx = fmaxf(px + 0.5f * pw, gx + 0.5f * gw);
            float my = fminf(py - 0.5f * ph, gy - 0.5f * gh);
            float My = fmaxf(py + 0.5f * ph, gy + 0.5f * gh);
            float cw = pw + gw - (Mx - mx);
            float ch = ph + gh - (My - my);
            float inter = (cw > 0.0f && ch > 0.0f) ? cw * ch : 0.0f;
            float uni = pa + gw * gh - inter;
            cur = fmaxf(cur, inter / uni);
        }
        bool ignore = cur > IGNORE_THRESH;
        int tIdx = cellTarg[((b * NA + a) * NH + j) * NW + i];
        if (tIdx >= 0) {
            // positive cell: tconf==1, coord + class losses
            const float* td = targData + ((size_t)b * T + tIdx) * 8;
            float cs = td[4];
            loss += cs * (softplusf(lx) - td[0] * lx);               // bce(lx,tx)
            loss += cs * (softplusf(ly) - td[1] * ly);               // bce(ly,ty)
            float dw = lw - td[2], dh = lh - td[3];
            loss += 0.5f * cs * (dw * dw + dh * dh);                 // mse/2
            loss += softplusf(lconf) - lconf;                        // bce(lconf,1)
            int cls = (int)td[5];
            if (ignore) {   // ign: whole class row was zeroed, cls entry = 1
                for (int c = 0; c < NC; ++c) {
                    float l = base[(5 + c) * HW];
                    loss += softplusf(l) - ((c == cls) ? l : 0.0f);
                }
            } else {        // only the cls entry is != -1
                float l = base[(5 + cls) * HW];
                loss += softplusf(l) - l;
            }
        } else if (!ignore) {
            loss += softplusf(lconf);                                 // bce(lconf,0)
        } // else: tconf == -1 -> excluded
    }
    float total = block_reduce_256(loss, sRed, tid);
    if (tid == 0) partials[blockIdx.x] = total;
}

__global__ __launch_bounds__(256) void k_final(const float* __restrict__ partials, int n,
                                               float* __restrict__ out) {
    __shared__ float sRed[256];
    int tid = threadIdx.x;
    float s = 0.0f;
    for (int k = tid; k < n; k += 256) s += partials[k];
    float total = block_reduce_256(s, sRed, tid);
    if (tid == 0) out[0] = total;
}

extern "C" void kernel_launch(void* const* d_in, const int* in_sizes, int n_in,
                              void* d_out, int out_size, void* d_ws, size_t ws_size,
                              hipStream_t stream) {
    const float* inp = (const float*)d_in[0];
    const float* tgt = (const float*)d_in[1];
    int nB = in_sizes[0] / (NB_CH * HW);          // 32
    int T = in_sizes[1] / (nB * 5);               // 50

    char* ws = (char*)d_ws;
    int nCells = nB * NA * HW;
    size_t off = 0;
    int* cellTarg = (int*)(ws + off);
    off += (((size_t)nCells * 4) + 255) & ~(size_t)255;
    int* nValid = (int*)(ws + off);
    off += (((size_t)nB * 4) + 255) & ~(size_t)255;
    float* gtBox = (float*)(ws + off);
    off += (((size_t)nB * T * 4 * 4) + 255) & ~(size_t)255;
    float* targData = (float*)(ws + off);
    off += (((size_t)nB * T * 8 * 4) + 255) & ~(size_t)255;
    float* partials = (float*)(ws + off);

    const int blocksPerB = (NA * HW + 255) / 256;   // 32
    int nBlocks = nB * blocksPerB;                  // 1024

    k_init<<<(nCells + 255) / 256, 256, 0, stream>>>(cellTarg, nCells);
    k_prep<<<nB, 256, 0, stream>>>(tgt, T, cellTarg, nValid, gtBox, targData);
    k_main<<<nBlocks, 256, 0, stream>>>(inp, T, cellTarg, nValid, gtBox, targData, partials);
    k_final<<<1, 256, 0, stream>>>(partials, nBlocks, (float*)d_out);
}